// ProbsNet_79293686218974
// MI455X (gfx1250) — compile-verified
//
#include <hip/hip_runtime.h>
#include <hip/hip_bf16.h>

typedef __attribute__((ext_vector_type(2))) float v2f;
typedef __attribute__((ext_vector_type(8))) float v8f;

#define NROWS 84
#define S_LEN 131072
#define CHUNKS 16
#define TPB 256
#define VEC4_PER_ROW   (S_LEN / 4)             // 32768 float4 per row
#define VEC4_PER_CHUNK (VEC4_PER_ROW / CHUNKS) // 2048 float4 per block
#define ITERS          (VEC4_PER_CHUNK / TPB)  // 8 per thread

__device__ __forceinline__ float fast_sigmoid(float z) {
  // sigmoid(z) = 1 / (1 + exp(-z)); exp(-z) = exp2(-z * log2(e))
  float e = __builtin_amdgcn_exp2f(-z * 1.44269504088896340736f);
  return __builtin_amdgcn_rcpf(1.0f + e);
}

// Streaming reduction: partial[bid] = sum over chunk of sigmoid(B*(bias+st))*w
__global__ void probsnet_stream(const float* __restrict__ ST0,
                                const float* __restrict__ W0,
                                const float* __restrict__ ST1,
                                const float* __restrict__ W1,
                                const float* __restrict__ BEV,
                                const float* __restrict__ BEV_p,
                                const float* __restrict__ Bs,
                                float* __restrict__ partials) {
  const int bid   = blockIdx.x;                 // [0, 2*NROWS*CHUNKS)
  const int pair  = bid / (NROWS * CHUNKS);
  const int rem   = bid - pair * (NROWS * CHUNKS);
  const int row   = rem / CHUNKS;
  const int chunk = rem - row * CHUNKS;

  const float bias = fmaxf(BEV_p[0], 0.0f) * BEV[0];
  const float B    = Bs[0];

  const float4* st = (const float4*)(pair ? ST1 : ST0);
  const float4* w  = (const float4*)(pair ? W1  : W0);
  const size_t base = (size_t)row * VEC4_PER_ROW + (size_t)chunk * VEC4_PER_CHUNK;

  float acc = 0.0f;
#pragma unroll
  for (int it = 0; it < ITERS; ++it) {
    size_t idx = base + (size_t)it * TPB + threadIdx.x;
    float4 s = st[idx];   // global_load_b128, coalesced
    float4 v = w[idx];
    acc += fast_sigmoid(B * (bias + s.x)) * v.x;
    acc += fast_sigmoid(B * (bias + s.y)) * v.y;
    acc += fast_sigmoid(B * (bias + s.z)) * v.z;
    acc += fast_sigmoid(B * (bias + s.w)) * v.w;
  }

  __shared__ float red[TPB];
  red[threadIdx.x] = acc;
  __syncthreads();
#pragma unroll
  for (int off = TPB / 2; off > 0; off >>= 1) {
    if (threadIdx.x < off) red[threadIdx.x] += red[threadIdx.x + off];
    __syncthreads();
  }
  if (threadIdx.x == 0) partials[bid] = red[0];
}

// Single wave32: sum partials, build zero-padded 16-row LDS tiles, then
// D = A(16x4) x B(4x16) via v_wmma_f32_16x16x4_f32 over 21 K-chunks of 84.
//   A rows 0..4 = P0..P4 (combination vectors), rows 5..15 = 0
//   B col 0 = tmp0, col 1 = tmp1, cols 2..15 = 0
__global__ void probsnet_finalize(const float* __restrict__ partials,
                                  const float* __restrict__ l0,
                                  const float* __restrict__ l1,
                                  const float* __restrict__ l2,
                                  const float* __restrict__ l3,
                                  const float* __restrict__ l4,
                                  float* __restrict__ out) {
  // Row length 84 floats = 336 B = multiple of 8 -> every (k0+kk) even index
  // is 8-byte aligned: lanes can use unconditional ds_load_b64.
  __shared__ float Pm[16][NROWS];   // A-matrix rows (zero-padded)
  __shared__ float Bm[16][NROWS];   // B-matrix "rows" indexed by column n
  __shared__ float dred[2][8];

  const int lane = threadIdx.x;  // 32 threads, one wave

  // Zero both tiles (16*84*2 = 2688 floats).
  for (int t = lane; t < 16 * NROWS; t += 32) {
    Pm[0][t] = 0.0f;
    Bm[0][t] = 0.0f;
  }
  __syncthreads();

  // Fill B rows 0/1: tmp[pair][row] = sum of the 16 per-chunk partials.
  for (int r = lane; r < 2 * NROWS; r += 32) {
    float s = 0.0f;
#pragma unroll
    for (int c = 0; c < CHUNKS; ++c) s += partials[r * CHUNKS + c];
    Bm[r / NROWS][r % NROWS] = s;
  }

  // Lanes 0..4: softmax over one 4-logit vector, expand to 84-entry combo row.
  if (lane < 5) {
    const float* L = (lane == 0) ? l0 : (lane == 1) ? l1 : (lane == 2) ? l2
                   : (lane == 3) ? l3 : l4;
    float x0 = L[0], x1 = L[1], x2 = L[2], x3 = L[3];
    float mx = fmaxf(fmaxf(x0, x1), fmaxf(x2, x3));
    float p[4];
    p[0] = __builtin_amdgcn_exp2f((x0 - mx) * 1.44269504088896340736f);
    p[1] = __builtin_amdgcn_exp2f((x1 - mx) * 1.44269504088896340736f);
    p[2] = __builtin_amdgcn_exp2f((x2 - mx) * 1.44269504088896340736f);
    p[3] = __builtin_amdgcn_exp2f((x3 - mx) * 1.44269504088896340736f);
    float inv = 1.0f / (p[0] + p[1] + p[2] + p[3]);
    p[0] *= inv; p[1] *= inv; p[2] *= inv; p[3] *= inv;
    int t = 0;
    for (int i = 0; i < 4; ++i) {
      Pm[lane][t++] = p[i];
      for (int j = 0; j < 4; ++j) {
        float pij = p[i] * p[j];
        Pm[lane][t++] = pij;
        for (int k = 0; k < 4; ++k) Pm[lane][t++] = pij * p[k];
      }
    }
  }
  __syncthreads();

  // WMMA fragment layout (32-bit, 16x16x4):
  //   A (16x4): lanes 0-15 -> M=lane, K={k0,k0+1}; lanes 16-31 -> K={k0+2,k0+3}
  //   B (4x16): lanes 0-15 -> N=lane, K rows {k0,k0+1}; lanes 16-31 -> {k0+2,k0+3}
  v8f cacc = {};
  const int mn = lane & 15;
  const int kk = (lane >> 4) * 2;
  const float* arow = &Pm[mn][0];
  const float* brow = &Bm[mn][0];
#pragma unroll 1
  for (int k0 = 0; k0 < NROWS; k0 += 4) {
    v2f a = *(const v2f*)(arow + k0 + kk);   // ds_load_b64, unconditional
    v2f b = *(const v2f*)(brow + k0 + kk);   // ds_load_b64, unconditional
    cacc = __builtin_amdgcn_wmma_f32_16x16x4_f32(
        /*neg_a=*/false, a, /*neg_b=*/false, b,
        /*c_mod=*/(short)0, cacc, /*reuse_a=*/false, /*reuse_b=*/false);
  }

  // D readout: lane 0 VGPR0 = D[0][0] = P0.tmp0; lane 1 VGPR m = D[m][1] = Pm.tmp1
  if (lane < 2) {
#pragma unroll
    for (int r = 0; r < 8; ++r) dred[lane][r] = cacc[r];
  }
  __syncthreads();
  if (lane == 0) {
    float mean = (dred[0][0] + dred[1][1] + dred[1][2] + dred[1][3] + dred[1][4]) * 0.2f;
    out[0] = mean;
  }
}

extern "C" void kernel_launch(void* const* d_in, const int* in_sizes, int n_in,
                              void* d_out, int out_size, void* d_ws, size_t ws_size,
                              hipStream_t stream) {
  const float* BEV   = (const float*)d_in[0];
  const float* ST0   = (const float*)d_in[1];
  const float* W0    = (const float*)d_in[2];
  const float* ST1   = (const float*)d_in[3];
  const float* W1    = (const float*)d_in[4];
  const float* p0    = (const float*)d_in[5];
  const float* p1    = (const float*)d_in[6];
  const float* p2    = (const float*)d_in[7];
  const float* p3    = (const float*)d_in[8];
  const float* p4    = (const float*)d_in[9];
  const float* BEV_p = (const float*)d_in[10];
  const float* B     = (const float*)d_in[11];

  float* partials = (float*)d_ws;            // 2*84*16 = 2688 floats
  float* out      = (float*)d_out;

  probsnet_stream<<<dim3(2 * NROWS * CHUNKS), dim3(TPB), 0, stream>>>(
      ST0, W0, ST1, W1, BEV, BEV_p, B, partials);
  probsnet_finalize<<<dim3(1), dim3(32), 0, stream>>>(
      partials, p0, p1, p2, p3, p4, out);
}